// _EMOptFunc_22342419873927
// MI455X (gfx1250) — compile-verified
//
#include <hip/hip_runtime.h>
#include <math.h>

#define KD 5
#define NSLOT 18                 // 0:w  1:w*(-logpdf)  2:w*klb  3..17: w*post_mean_sq (upper tri)
#define NTHR 256
#define F_K_LOG2PI (5.0f * 1.8378770664093453f)

// workspace layout (floats)
#define WS_PINV 0      // 25: prior_inv (full symmetric)
#define WS_LDP  25     // 1:  logdet(prior)
#define WS_GMAX 26     // 1:  global max logit
#define WS_BMAX 32     // nblk: per-block max; partials [nblk][NSLOT] follow

typedef __attribute__((ext_vector_type(2))) float v2f;
typedef __attribute__((ext_vector_type(8))) float v8f;

// ---------- 5x5 SPD helpers (fully unrolled -> VGPRs) ----------
__device__ __forceinline__ float chol5(const float A[KD][KD], float L[KD][KD]) {
  float ld = 0.0f;
#pragma unroll
  for (int j = 0; j < KD; ++j) {
    float s = A[j][j];
#pragma unroll
    for (int k = 0; k < KD; ++k) if (k < j) s -= L[j][k] * L[j][k];
    float ljj = sqrtf(s);
    L[j][j] = ljj;
    ld += logf(ljj);
    float inv = 1.0f / ljj;
#pragma unroll
    for (int i = 0; i < KD; ++i) if (i > j) {
      float t = A[i][j];
#pragma unroll
      for (int k = 0; k < KD; ++k) if (k < j) t -= L[i][k] * L[j][k];
      L[i][j] = t * inv;
    }
  }
  return 2.0f * ld;  // logdet(A)
}

__device__ __forceinline__ void fsolve5(const float L[KD][KD], const float r[KD], float y[KD]) {
#pragma unroll
  for (int i = 0; i < KD; ++i) {
    float t = r[i];
#pragma unroll
    for (int k = 0; k < KD; ++k) if (k < i) t -= L[i][k] * y[k];
    y[i] = t / L[i][i];
  }
}

__device__ __forceinline__ void bsolve5(const float L[KD][KD], const float y[KD], float m[KD]) {
#pragma unroll
  for (int i = KD - 1; i >= 0; --i) {
    float t = y[i];
#pragma unroll
    for (int k = 0; k < KD; ++k) if (k > i) t -= L[k][i] * m[k];
    m[i] = t / L[i][i];
  }
}

__device__ __forceinline__ void inv_from_chol5(const float L[KD][KD], float C[KD][KD]) {
  float Li[KD][KD];
#pragma unroll
  for (int j = 0; j < KD; ++j) {
    Li[j][j] = 1.0f / L[j][j];
#pragma unroll
    for (int i = 0; i < KD; ++i) if (i > j) {
      float t = 0.0f;
#pragma unroll
      for (int k = 0; k < KD; ++k) if (k >= j && k < i) t += L[i][k] * Li[k][j];
      Li[i][j] = -t / L[i][i];
    }
  }
#pragma unroll
  for (int i = 0; i < KD; ++i)
#pragma unroll
    for (int j = 0; j < KD; ++j) {
      float t = 0.0f;
#pragma unroll
      for (int k = 0; k < KD; ++k) if (k >= i && k >= j) t += Li[k][i] * Li[k][j];
      C[i][j] = t;
    }
}

__device__ __forceinline__ float wave_max32(float v) {
#pragma unroll
  for (int o = 16; o > 0; o >>= 1) v = fmaxf(v, __shfl_down(v, o, 32));
  return v;
}
__device__ __forceinline__ float wave_sum32(float v) {
#pragma unroll
  for (int o = 16; o > 0; o >>= 1) v += __shfl_down(v, o, 32);
  return v;
}

// ---------- kernels ----------
__global__ void k_prior(const float* __restrict__ ec, const int* __restrict__ li,
                        float* __restrict__ ws) {
  if (blockIdx.x == 0 && threadIdx.x == 0) {
    const float* A0 = ec + (size_t)li[0] * (KD * KD);
    float A[KD][KD], L[KD][KD], C[KD][KD];
#pragma unroll
    for (int i = 0; i < KD; ++i)
#pragma unroll
      for (int j = 0; j < KD; ++j) A[i][j] = A0[i * KD + j];
    float ld = chol5(A, L);
    inv_from_chol5(L, C);
#pragma unroll
    for (int i = 0; i < KD; ++i)
#pragma unroll
      for (int j = 0; j < KD; ++j) ws[WS_PINV + i * KD + j] = C[i][j];
    ws[WS_LDP] = ld;
  }
}

__global__ void k_pass1(const float* __restrict__ beta, const float* __restrict__ shat2,
                        const float* __restrict__ pi, const float* __restrict__ ws,
                        float* __restrict__ bmax, int P) {
  float Pinv[KD][KD];
#pragma unroll
  for (int i = 0; i < KD; ++i)
#pragma unroll
    for (int j = 0; j < KD; ++j) Pinv[i][j] = ws[WS_PINV + i * KD + j];

  const int stride = gridDim.x * blockDim.x;
  float mx = -INFINITY;
  for (int p = blockIdx.x * blockDim.x + threadIdx.x; p < P; p += stride) {
    __builtin_prefetch(shat2 + (size_t)(p + stride) * 25, 0, 1);
    float d[KD], r[KD];
#pragma unroll
    for (int i = 0; i < KD; ++i) d[i] = shat2[(size_t)p * 25 + i * 6];  // diag only
#pragma unroll
    for (int i = 0; i < KD; ++i) r[i] = beta[(size_t)p * KD + i] / d[i];
    float A[KD][KD];
#pragma unroll
    for (int i = 0; i < KD; ++i)
#pragma unroll
      for (int j = 0; j < KD; ++j) A[i][j] = Pinv[i][j] + ((i == j) ? 1.0f / d[i] : 0.0f);
    float L[KD][KD];
    float ldA = chol5(A, L);           // logdet(post_precision) = -ld_post
    float y[KD];
    fsolve5(L, r, y);                  // maha = ||L^-1 r||^2
    float maha = 0.0f;
#pragma unroll
    for (int i = 0; i < KD; ++i) maha += y[i] * y[i];
    float logit = logf(pi[p]) + 0.5f * (F_K_LOG2PI - ldA + maha);  // log pi - logpdf
    mx = fmaxf(mx, logit);
  }
  __shared__ float sred[NTHR / 32];
  float wmx = wave_max32(mx);
  int wid = threadIdx.x >> 5, lid = threadIdx.x & 31;
  if (lid == 0) sred[wid] = wmx;
  __syncthreads();
  if (threadIdx.x == 0) {
    float m = sred[0];
#pragma unroll
    for (int w = 1; w < NTHR / 32; ++w) m = fmaxf(m, sred[w]);
    bmax[blockIdx.x] = m;
  }
}

__global__ void k_gmax(float* __restrict__ ws, const float* __restrict__ bmax, int nblk) {
  float m = -INFINITY;
  for (int i = threadIdx.x; i < nblk; i += blockDim.x) m = fmaxf(m, bmax[i]);
  __shared__ float sred[NTHR / 32];
  float wmx = wave_max32(m);
  int wid = threadIdx.x >> 5, lid = threadIdx.x & 31;
  if (lid == 0) sred[wid] = wmx;
  __syncthreads();
  if (threadIdx.x == 0) {
    float g = sred[0];
#pragma unroll
    for (int w = 1; w < NTHR / 32; ++w) g = fmaxf(g, sred[w]);
    ws[WS_GMAX] = g;
  }
}

__global__ void k_pass2(const float* __restrict__ beta, const float* __restrict__ shat2,
                        const float* __restrict__ pi, const float* __restrict__ ws,
                        float* __restrict__ part, int P) {
  float Pinv[KD][KD];
#pragma unroll
  for (int i = 0; i < KD; ++i)
#pragma unroll
    for (int j = 0; j < KD; ++j) Pinv[i][j] = ws[WS_PINV + i * KD + j];
  const float ldp  = ws[WS_LDP];
  const float gmax = ws[WS_GMAX];

  float acc[NSLOT];
#pragma unroll
  for (int s = 0; s < NSLOT; ++s) acc[s] = 0.0f;

  const int stride = gridDim.x * blockDim.x;
  for (int p = blockIdx.x * blockDim.x + threadIdx.x; p < P; p += stride) {
    __builtin_prefetch(shat2 + (size_t)(p + stride) * 25, 0, 1);
    float d[KD], r[KD];
#pragma unroll
    for (int i = 0; i < KD; ++i) d[i] = shat2[(size_t)p * 25 + i * 6];
#pragma unroll
    for (int i = 0; i < KD; ++i) r[i] = beta[(size_t)p * KD + i] / d[i];
    float A[KD][KD];
#pragma unroll
    for (int i = 0; i < KD; ++i)
#pragma unroll
      for (int j = 0; j < KD; ++j) A[i][j] = Pinv[i][j] + ((i == j) ? 1.0f / d[i] : 0.0f);
    float L[KD][KD];
    float ldA = chol5(A, L);
    float y[KD], m[KD];
    fsolve5(L, r, y);
    bsolve5(L, y, m);                  // post_mean
    float maha = 0.0f;
#pragma unroll
    for (int i = 0; i < KD; ++i) maha += y[i] * y[i];
    float lpi = logf(pi[p]);
    float logit = lpi + 0.5f * (F_K_LOG2PI - ldA + maha);

    float C[KD][KD];
    inv_from_chol5(L, C);              // post_covar
    float p1 = -(float)KD, p2 = 0.0f;
#pragma unroll
    for (int i = 0; i < KD; ++i) {
      float ti = 0.0f;
#pragma unroll
      for (int j = 0; j < KD; ++j) { p1 += Pinv[i][j] * C[i][j]; ti += Pinv[i][j] * m[j]; }
      p2 += m[i] * ti;
    }
    float klb = 0.5f * (p1 + p2 + (ldp + ldA));   // p3 = ld_prior - ld_post = ldp + ldA
    float w = expf(logit - gmax);
    acc[0] += w;
    acc[1] += w * (logit - lpi);       // w * (-logpdf)
    acc[2] += w * klb;
    int u = 3;
#pragma unroll
    for (int i = 0; i < KD; ++i)
#pragma unroll
      for (int j = 0; j < KD; ++j) if (j >= i) { acc[u] += w * (C[i][j] + m[i] * m[j]); ++u; }
  }

  __shared__ float sred[NTHR / 32];
  int wid = threadIdx.x >> 5, lid = threadIdx.x & 31;
#pragma unroll
  for (int s = 0; s < NSLOT; ++s) {
    float v = wave_sum32(acc[s]);
    if (lid == 0) sred[wid] = v;
    __syncthreads();
    if (threadIdx.x == 0) {
      float t = 0.0f;
#pragma unroll
      for (int w = 0; w < NTHR / 32; ++w) t += sred[w];
      part[(size_t)blockIdx.x * NSLOT + s] = t;
    }
    __syncthreads();
  }
}

// Final reduction of part[nblk][NSLOT] with the f32 matrix core (B = ones -> row sums),
// then compute KL terms and write the 251 outputs. One wave (EXEC all-ones for WMMA).
__global__ void k_final(const float* __restrict__ ec, const int* __restrict__ li,
                        const float* __restrict__ ws, const float* __restrict__ part,
                        float* __restrict__ out, int nblk, int ecn) {
  __shared__ float ssum[NSLOT];
  const int lane = threadIdx.x;

#if defined(__gfx1250__) && __has_builtin(__builtin_amdgcn_wmma_f32_16x16x4_f32)
  // A(M,K) layout: lane = M + 16*(K>=2), vgpr = K&1. With B=ones: D[M][*] += sum_K A[M][K].
  v8f c1 = {0.f, 0.f, 0.f, 0.f, 0.f, 0.f, 0.f, 0.f};
  v8f c2 = {0.f, 0.f, 0.f, 0.f, 0.f, 0.f, 0.f, 0.f};
  v2f bones; bones.x = 1.0f; bones.y = 1.0f;
  const int s1 = lane & 15;
  const int base = (lane >= 16) ? 2 : 0;
  const int T = (nblk + 3) >> 2;
  for (int t = 0; t < T; ++t) {
    int b0 = 4 * t + base, b1 = b0 + 1;
    v2f a1, a2;
    a1.x = (b0 < nblk) ? part[(size_t)b0 * NSLOT + s1] : 0.0f;
    a1.y = (b1 < nblk) ? part[(size_t)b1 * NSLOT + s1] : 0.0f;
    float z0 = 0.0f, z1 = 0.0f;
    if (s1 < NSLOT - 16) {  // slots 16,17 in second accumulator, rows M=0,1
      z0 = (b0 < nblk) ? part[(size_t)b0 * NSLOT + 16 + s1] : 0.0f;
      z1 = (b1 < nblk) ? part[(size_t)b1 * NSLOT + 16 + s1] : 0.0f;
    }
    a2.x = z0; a2.y = z1;
    c1 = __builtin_amdgcn_wmma_f32_16x16x4_f32(false, a1, false, bones, (short)0, c1, false, false);
    c2 = __builtin_amdgcn_wmma_f32_16x16x4_f32(false, a2, false, bones, (short)0, c2, false, false);
  }
  // D layout: vgpr j, lanes 0-15 -> (M=j, N=lane); lanes 16-31 -> (M=8+j, N=lane-16).
  if (lane == 0) {
#pragma unroll
    for (int j = 0; j < 8; ++j) ssum[j] = c1[j];
    ssum[16] = c2[0];
    ssum[17] = c2[1];
  }
  if (lane == 16) {
#pragma unroll
    for (int j = 0; j < 8; ++j) ssum[8 + j] = c1[j];
  }
#else
  if (lane == 0) {
    for (int s = 0; s < NSLOT; ++s) {
      float t = 0.0f;
      for (int b = 0; b < nblk; ++b) t += part[(size_t)b * NSLOT + s];
      ssum[s] = t;
    }
  }
#endif
  __syncthreads();

  const int l = li[0];
  const int lo = l * KD * KD, hi = lo + KD * KD;
  for (int idx = lane; idx < ecn; idx += 32)
    if (idx < lo || idx >= hi) out[idx] = ec[idx];

  if (lane == 0) {
    float Wsum = ssum[0];
    float invW = 1.0f / Wsum;
    float gmax = ws[WS_GMAX];
    float kl_alpha = ssum[1] * invW - (gmax + logf(Wsum));
    float kl = kl_alpha + ssum[2] * invW;
    float S[KD][KD];
    int u = 3;
#pragma unroll
    for (int i = 0; i < KD; ++i)
#pragma unroll
      for (int j = 0; j < KD; ++j) if (j >= i) { float v = ssum[u] * invW; S[i][j] = v; S[j][i] = v; ++u; }
#pragma unroll
    for (int i = 0; i < KD; ++i)
#pragma unroll
      for (int j = 0; j < KD; ++j) out[lo + i * KD + j] = S[i][j];
    out[ecn] = kl;
  }
}

extern "C" void kernel_launch(void* const* d_in, const int* in_sizes, int n_in,
                              void* d_out, int out_size, void* d_ws, size_t ws_size,
                              hipStream_t stream) {
  (void)n_in; (void)out_size;
  const float* beta  = (const float*)d_in[0];
  const float* shat2 = (const float*)d_in[1];
  // d_in[2] (inv_shat2) deliberately unread: diag == 1/d, recomputed in-kernel (saves ~100 MB traffic)
  const float* pi    = (const float*)d_in[3];
  const float* ec    = (const float*)d_in[4];
  const int*   li    = (const int*)d_in[5];
  const int P   = in_sizes[3];
  const int ecn = in_sizes[4];
  float* ws  = (float*)d_ws;
  float* out = (float*)d_out;

  int nblk = 1024;
  while (nblk > 1 &&
         ((size_t)WS_BMAX + (size_t)nblk * (1 + NSLOT)) * sizeof(float) > ws_size)
    nblk >>= 1;

  float* bmax = ws + WS_BMAX;
  float* part = ws + WS_BMAX + nblk;

  k_prior<<<1, 32, 0, stream>>>(ec, li, ws);
  k_pass1<<<nblk, NTHR, 0, stream>>>(beta, shat2, pi, ws, bmax, P);
  k_gmax<<<1, NTHR, 0, stream>>>(ws, bmax, nblk);
  k_pass2<<<nblk, NTHR, 0, stream>>>(beta, shat2, pi, ws, part, P);
  k_final<<<1, 32, 0, stream>>>(ec, li, ws, part, out, nblk, ecn);
}